// RNN1_7980049236090
// MI455X (gfx1250) — compile-verified
//
#include <hip/hip_runtime.h>
#include <hip/hip_bf16.h>
#include <stdint.h>

// ---------------------------------------------------------------------------
// RNN (B=16384, T=256, I=5, H=2) + MLP head (2->128 ReLU -> 1), fused.
// MI455X / gfx1250 design:
//   * TDM (tensor_load_to_lds) streams 32-row x-tiles (32 batch x 32 steps,
//     160 dwords/row, row stride 1280 dwords) into LDS, double buffered,
//     synced with s_wait_tensorcnt. TDM padding (2 dwords per 32) makes the
//     LDS row stride 170 dwords -> gcd(170,64)=2 -> conflict-free reads.
//   * 512 blocks x 32 threads (one wave32 per block): no barriers needed.
//   * Recurrence uses v_tanh_f32 (CDNA5 transcendental).
//   * MLP head uses V_WMMA_F32_16X16X4_F32 (K=2 padded to 4), 2 M-tiles x
//     8 N-tiles per wave, shfl_xor reduction over 16-lane halves.
// ---------------------------------------------------------------------------

typedef float        v2f   __attribute__((ext_vector_type(2)));
typedef float        v8f   __attribute__((ext_vector_type(8)));
typedef unsigned int u32x4 __attribute__((ext_vector_type(4)));
typedef int          i32x4 __attribute__((ext_vector_type(4)));
typedef int          i32x8 __attribute__((ext_vector_type(8)));

#define B_TOT   16384
#define T_LEN   256
#define I_DIM   5
#define F_DIM   128
#define TCHUNK  32
#define NCHUNK  (T_LEN / TCHUNK)        // 8
#define ROW_RAW (TCHUNK * I_DIM)        // 160 dwords of payload per row/chunk
#define ROW_DW  170                     // 160 + 5 pads * 2 dwords (TDM pad)
#define BUF_DW  (32 * ROW_DW)           // 5440 dwords = 21760 B per buffer

// padded intra-row dword index: TDM inserts 2 pad dwords after every 32
#define PIDX(j) ((j) + 2 * ((j) >> 5))

__device__ __forceinline__ float hw_tanh(float v) {
#if __has_builtin(__builtin_amdgcn_tanhf)
  return __builtin_amdgcn_tanhf(v);        // v_tanh_f32
#else
  return tanhf(v);
#endif
}

// Issue one 2D TDM tile load: 32 rows x 160 dwords, row stride 1280 dwords.
__device__ __forceinline__ void tdm_load_tile(const float* gsrc, unsigned lds_byte) {
  unsigned long long ga = (unsigned long long)(uintptr_t)gsrc;
  u32x4 g0;
  g0.x = 1u;                                                  // count=1 (valid user D#)
  g0.y = lds_byte;                                            // lds_addr (bytes)
  g0.z = (unsigned)(ga & 0xFFFFFFFFull);                      // global_addr[31:0]
  g0.w = (unsigned)((ga >> 32) & 0x1FFFFFFull) | (2u << 30);  // global_addr[56:32], type=2
  i32x8 g1;
  // data_size=2 (4B) | pad_enable | pad_interval=4 (32 dwords) | pad_amount=1 (2 dwords)
  g1.s0 = (int)((2u << 16) | (1u << 20) | (4u << 22) | (1u << 25));
  g1.s1 = (int)((unsigned)ROW_RAW << 16);   // tensor_dim0[15:0] = 160
  g1.s2 = (int)(32u << 16);                 // dim0 hi = 0 | tensor_dim1 = 32
  g1.s3 = (int)((unsigned)ROW_RAW << 16);   // dim1 hi = 0 | tile_dim0 = 160
  g1.s4 = 32;                               // tile_dim1 = 32, tile_dim2 = 0
  g1.s5 = T_LEN * I_DIM;                    // tensor_dim0_stride = 1280 elems
  g1.s6 = 0;
  g1.s7 = 0;
  i32x4 z4 = {0, 0, 0, 0};
  i32x8 z8 = {0, 0, 0, 0, 0, 0, 0, 0};
  // clang-23 / therock form: 6 args (g0, g1, g2, g3, extra, cpol)
  __builtin_amdgcn_tensor_load_to_lds(g0, g1, z4, z4, z8, 0);
}

__global__ __launch_bounds__(32)
void rnn_fused_kernel(const float* __restrict__ x,
                      const float* __restrict__ W_ih, const float* __restrict__ W_hh,
                      const float* __restrict__ b_ih, const float* __restrict__ b_hh,
                      const float* __restrict__ W1,  const float* __restrict__ b1,
                      const float* __restrict__ W2,  const float* __restrict__ b2,
                      float* __restrict__ out) {
  __shared__ float xbuf[2][BUF_DW];

  const int lane  = threadIdx.x;            // 0..31, one batch row per lane
  const int bbase = blockIdx.x * 32;        // this block's batch base

  // --- uniform weight preload (scalar loads) ---
  float wih[2][I_DIM];
#pragma unroll
  for (int i = 0; i < I_DIM; ++i) { wih[0][i] = W_ih[i]; wih[1][i] = W_ih[I_DIM + i]; }
  const float whh00 = W_hh[0], whh01 = W_hh[1], whh10 = W_hh[2], whh11 = W_hh[3];
  const float bs0 = b_ih[0] + b_hh[0];
  const float bs1 = b_ih[1] + b_hh[1];

  const float* xblk = x + (size_t)bbase * (T_LEN * I_DIM);
  const unsigned lds0 = (unsigned)(uintptr_t)(void*)&xbuf[0][0]; // flat->LDS: low 32 bits
  const unsigned lds1 = (unsigned)(uintptr_t)(void*)&xbuf[1][0];

  // prologue: chunk 0 into buffer 0
  tdm_load_tile(xblk, lds0);

  float h0 = 0.f, h1 = 0.f;
  for (int c = 0; c < NCHUNK; ++c) {
    if (c + 1 < NCHUNK) {
      tdm_load_tile(xblk + (c + 1) * (TCHUNK * I_DIM), ((c + 1) & 1) ? lds1 : lds0);
      __builtin_amdgcn_s_wait_tensorcnt(1);   // oldest (chunk c) has landed
    } else {
      __builtin_amdgcn_s_wait_tensorcnt(0);
    }
    asm volatile("" ::: "memory");            // keep LDS reads after the wait

    const float* lrow = &xbuf[c & 1][lane * ROW_DW];
#pragma unroll
    for (int tt = 0; tt < TCHUNK; ++tt) {
      const int j = tt * I_DIM;
      const float x0 = lrow[PIDX(j + 0)];
      const float x1 = lrow[PIDX(j + 1)];
      const float x2 = lrow[PIDX(j + 2)];
      const float x3 = lrow[PIDX(j + 3)];
      const float x4 = lrow[PIDX(j + 4)];
      float xp0 = bs0 + wih[0][0]*x0 + wih[0][1]*x1 + wih[0][2]*x2 + wih[0][3]*x3 + wih[0][4]*x4;
      float xp1 = bs1 + wih[1][0]*x0 + wih[1][1]*x1 + wih[1][2]*x2 + wih[1][3]*x3 + wih[1][4]*x4;
      const float a0 = xp0 + whh00 * h0 + whh01 * h1;
      const float a1 = xp1 + whh10 * h0 + whh11 * h1;
      h0 = hw_tanh(a0);
      h1 = hw_tanh(a1);
    }
  }

  // --- MLP head: relu(hn) -> W1/relu -> W2, via V_WMMA_F32_16X16X4_F32 ---
  const float r0 = fmaxf(h0, 0.f);
  const float r1 = fmaxf(h1, 0.f);
  const float r0u = __shfl_down(r0, 16);      // lanes<16 fetch upper half's h
  const float r1u = __shfl_down(r1, 16);
  const bool  lo  = lane < 16;
  const float msk = lo ? 1.f : 0.f;           // branch-free lane mask (keeps EXEC intact)

  // A (16x4 f32): lanes 0-15 hold K=0,1 of rows M=lane; lanes 16-31 (K=2,3) = 0
  v2f A0; A0.x = r0  * msk; A0.y = r1  * msk;   // batches bbase+0..15
  v2f A1; A1.x = r0u * msk; A1.y = r1u * msk;   // batches bbase+16..31

  const int nl = lane & 15;
  float acc0[8], acc1[8];
#pragma unroll
  for (int r = 0; r < 8; ++r) { acc0[r] = 0.f; acc1[r] = 0.f; }

#pragma unroll
  for (int nb = 0; nb < F_DIM; nb += 16) {
    const int n = nb + nl;
    // B (4x16 f32): lanes 0-15 hold K=0,1 for col N=lane; lanes 16-31 (K=2,3) = 0.
    // Unconditional packed b64 load on all lanes; zero upper half via mask mul.
    const v2f w1v = *(const v2f*)(W1 + 2 * n);
    v2f Bv; Bv.x = w1v.x * msk; Bv.y = w1v.y * msk;
    v8f C = {0.f, 0.f, 0.f, 0.f, 0.f, 0.f, 0.f, 0.f};
    v8f d0 = __builtin_amdgcn_wmma_f32_16x16x4_f32(false, A0, false, Bv, (short)0, C, false, false);
    v8f d1 = __builtin_amdgcn_wmma_f32_16x16x4_f32(false, A1, false, Bv, (short)0, C, false, false);
    const float b1n = b1[n];
    const float w2n = W2[n];
#pragma unroll
    for (int r = 0; r < 8; ++r) {
      acc0[r] += fmaxf(d0[r] + b1n, 0.f) * w2n;
      acc1[r] += fmaxf(d1[r] + b1n, 0.f) * w2n;
    }
  }

  // Reduce over the 16 lanes of each half (columns of W1/W2 are split there).
#pragma unroll
  for (int m = 1; m < 16; m <<= 1) {
#pragma unroll
    for (int r = 0; r < 8; ++r) {
      acc0[r] += __shfl_xor(acc0[r], m);
      acc1[r] += __shfl_xor(acc1[r], m);
    }
  }

  const float bias2 = b2[0];
  // D layout: VGPR r, lanes0-15 -> M=r ; lanes16-31 -> M=8+r
  if (lane == 0) {
#pragma unroll
    for (int r = 0; r < 8; ++r) {
      out[bbase + r]      = acc0[r] + bias2;   // batches 0..7
      out[bbase + 16 + r] = acc1[r] + bias2;   // batches 16..23
    }
  } else if (lane == 16) {
#pragma unroll
    for (int r = 0; r < 8; ++r) {
      out[bbase + 8 + r]  = acc0[r] + bias2;   // batches 8..15
      out[bbase + 24 + r] = acc1[r] + bias2;   // batches 24..31
    }
  }
}

extern "C" void kernel_launch(void* const* d_in, const int* in_sizes, int n_in,
                              void* d_out, int out_size, void* d_ws, size_t ws_size,
                              hipStream_t stream) {
  (void)in_sizes; (void)n_in; (void)out_size; (void)d_ws; (void)ws_size;
  const float* x    = (const float*)d_in[0];
  const float* W_ih = (const float*)d_in[1];
  const float* W_hh = (const float*)d_in[2];
  const float* b_ih = (const float*)d_in[3];
  const float* b_hh = (const float*)d_in[4];
  const float* W1   = (const float*)d_in[5];
  const float* b1   = (const float*)d_in[6];
  const float* W2   = (const float*)d_in[7];
  const float* b2   = (const float*)d_in[8];
  float* out = (float*)d_out;

  rnn_fused_kernel<<<B_TOT / 32, 32, 0, stream>>>(x, W_ih, W_hh, b_ih, b_hh,
                                                  W1, b1, W2, b2, out);
}